// WeightedHausdorffDistance_82729660055689
// MI455X (gfx1250) — compile-verified
//
#include <hip/hip_runtime.h>
#include <hip/hip_bf16.h>
#include <math.h>

// -------------------------------------------------------------------------
// WeightedHausdorffDistance on MI455X (gfx1250)
//
// Core trick: |x - y|^2 == dot(e, f) with e = [x0, x1, |x|^2, 1],
//                                      f = [-2y0, -2y1, 1, |y|^2]
// -> all-pairs squared distances map exactly onto V_WMMA_F32_16X16X4_F32
//    (A 16x4 f32, B 4x16 f32, C/D 16x16 f32), full fp32 precision.
// One wave handles one 16-pixel tile vs all 128 points (8 WMMAs).
// sqrt uses raw V_SQRT_F32 (TRANS op, ~1 ulp) instead of the IEEE libm
// expansion, which dominated the VALU count in the previous revision.
// -------------------------------------------------------------------------

typedef __attribute__((ext_vector_type(2))) float v2f;
typedef __attribute__((ext_vector_type(8))) float v8f;

#define HH 256
#define WW 256
#define NPIX (HH * WW)
#define BB 8
#define MM 128
#define MAXD 362.03867196751236f // sqrt(256^2 + 256^2)

// Workspace layout (floats):
//  [0..7]            t1num[b]   : sum p * min_m d      (atomicAdd)
//  [8..15]           psum[b]    : sum p                (atomicAdd)
//  [16..16+1023]     pmin[b*128+m] : min over pixels of weighted dist
//                    (float value, updated via uint atomicMin on the
//                     nonnegative-float bit pattern, which is monotone)
//  [1040]            focal numerator sum (pos_term + neg_term)
//  [1041]            pos count
#define WS_T1 0
#define WS_PS 8
#define WS_PM 16
#define WS_FA (16 + BB * MM)
#define WS_TOTAL (WS_FA + 2)

__global__ __launch_bounds__(256) void init_ws_kernel(float* ws) {
  int i = blockIdx.x * blockDim.x + threadIdx.x;
  if (i < WS_PM) {
    ws[i] = 0.0f;
  } else if (i < WS_FA) {
    ((unsigned int*)ws)[i] = 0x7F7FFFFFu; // FLT_MAX bits
  } else if (i < WS_TOTAL) {
    ws[i] = 0.0f;
  }
}

// One wave = one 16-pixel tile of one batch; 8 waves / block.
__global__ __launch_bounds__(256) void whd_wmma_kernel(
    const float* __restrict__ prob_map,  // [B, NPIX]
    const float* __restrict__ gt,        // [B, MM, 2]
    const float* __restrict__ orig,      // [B, 2]
    float* __restrict__ t1num,           // [B]
    float* __restrict__ psum,            // [B]
    unsigned int* __restrict__ pminbits) // [B, MM] (float bits)
{
  const int b    = blockIdx.y;
  const int lane = threadIdx.x & 31;
  const int wave = threadIdx.x >> 5;
  const int tile = blockIdx.x * 8 + wave; // 16-pixel tile, 0..4095
  const int half = lane >> 4;
  const int l15  = lane & 15;

  const float n0 = orig[b * 2 + 0] * (1.0f / (float)HH);
  const float n1 = orig[b * 2 + 1] * (1.0f / (float)WW);

  // ---- A fragment (pixels). 16x4 f32 A layout:
  //   lanes 0-15 : v0=K0, v1=K1 ; lanes 16-31 : v0=K2, v1=K3, M = lane&15
  const int   pix = tile * 16 + l15;
  const float rr  = (float)(pix >> 8);
  const float cc  = (float)(pix & 255);
  const float e0  = rr * n0;
  const float e1  = cc * n1;
  v2f a;
  if (half == 0) { a.x = e0;                a.y = e1;   }
  else           { a.x = e0 * e0 + e1 * e1; a.y = 1.0f; }

  // ---- B fragments (points), all 8 point-tiles preloaded.
  //   4x16 f32 B layout: lanes 0-15: v0=K0, v1=K1; lanes 16-31: v0=K2, v1=K3
  v2f bf[8];
#pragma unroll
  for (int pt = 0; pt < 8; ++pt) {
    const int   point = pt * 16 + l15;
    const float g0 = gt[(b * MM + point) * 2 + 0];
    const float g1 = gt[(b * MM + point) * 2 + 1];
    const float y0 = g0 * n0;
    const float y1 = g1 * n1;
    if (half == 0) { bf[pt].x = -2.0f * y0; bf[pt].y = -2.0f * y1;        }
    else           { bf[pt].x = 1.0f;       bf[pt].y = y0 * y0 + y1 * y1; }
  }

  // p for the 8 pixel rows covered by this half (M = half*8 + r)
  float pr[8];
#pragma unroll
  for (int r = 0; r < 8; ++r)
    pr[r] = prob_map[(size_t)b * NPIX + tile * 16 + half * 8 + r];

  float mind[8];
#pragma unroll
  for (int r = 0; r < 8; ++r) mind[r] = 3.0e38f;

#pragma unroll
  for (int pt = 0; pt < 8; ++pt) {
    v8f c = {};
    // D = A x B + 0  ->  c[r] = |pixel(half*8+r) - point(pt*16+l15)|^2
    c = __builtin_amdgcn_wmma_f32_16x16x4_f32(
        /*neg_a=*/false, a, /*neg_b=*/false, bf[pt],
        /*c_mod=*/(short)0, c, /*reuse_a=*/false, /*reuse_b=*/false);

    float wmin = 3.0e38f;
#pragma unroll
    for (int r = 0; r < 8; ++r) {
      // raw V_SQRT_F32 (TRANS): single instruction, ~1 ulp — ample here
      const float d = __builtin_amdgcn_sqrtf(fmaxf(c[r], 0.0f));
      mind[r] = fminf(mind[r], d);                        // term1: min over m
      const float w = fmaf(pr[r], d - MAXD, MAXD);        // (1-p)*MAXD + p*d
      wmin = fminf(wmin, w);
    }
    // column N=l15 lives in both halves (rows 0-7 vs 8-15): combine halves
    wmin = fminf(wmin, __shfl_xor(wmin, 16, 32));
    if (half == 0) {
      atomicMin(&pminbits[b * MM + pt * 16 + l15], __float_as_uint(wmin));
    }
  }

  // term1: reduce mind[r] over the 16 lanes of each half (min over points)
#pragma unroll
  for (int r = 0; r < 8; ++r) {
    float v = mind[r];
    v = fminf(v, __shfl_xor(v, 1, 32));
    v = fminf(v, __shfl_xor(v, 2, 32));
    v = fminf(v, __shfl_xor(v, 4, 32));
    v = fminf(v, __shfl_xor(v, 8, 32));
    mind[r] = v;
  }
  if (l15 == 0) { // lane 0 covers pixels tile*16+0..7, lane 16 covers +8..15
    float s = 0.0f, ps = 0.0f;
#pragma unroll
    for (int r = 0; r < 8; ++r) { s += pr[r] * mind[r]; ps += pr[r]; }
    atomicAdd(&t1num[b], s);
    atomicAdd(&psum[b], ps);
  }
}

// Center focal loss on the LAST batch element only (reference loop bug kept).
__global__ __launch_bounds__(256) void focal_kernel(
    const float* __restrict__ prob_map, const float* __restrict__ gt_map,
    float* __restrict__ facc /* [2]: term sum, pos count */)
{
  __shared__ float s_sum[256];
  __shared__ float s_pos[256];
  const int i = blockIdx.x * 256 + threadIdx.x;
  const float* __restrict__ pred_p = prob_map + (size_t)(BB - 1) * NPIX;
  const float* __restrict__ g_p    = gt_map   + (size_t)(BB - 1) * NPIX;

  float sum = 0.0f, pos = 0.0f;
  if (i < NPIX) {
    const float pred = pred_p[i];
    const float g    = g_p[i];
    if (g == 1.0f) {
      const float om = 1.0f - pred;
      sum = om * om * logf(pred);
      pos = 1.0f;
    } else {
      const float og  = 1.0f - g;
      const float og2 = og * og;
      sum = og2 * og2 * pred * pred * log1pf(-pred);
    }
  }
  s_sum[threadIdx.x] = sum;
  s_pos[threadIdx.x] = pos;
  __syncthreads();
  for (int s = 128; s > 0; s >>= 1) {
    if (threadIdx.x < s) {
      s_sum[threadIdx.x] += s_sum[threadIdx.x + s];
      s_pos[threadIdx.x] += s_pos[threadIdx.x + s];
    }
    __syncthreads();
  }
  if (threadIdx.x == 0) {
    atomicAdd(&facc[0], s_sum[0]);
    atomicAdd(&facc[1], s_pos[0]);
  }
}

__global__ __launch_bounds__(256) void finalize_kernel(
    const float* __restrict__ ws, float* __restrict__ out)
{
  const float* t1num = ws + WS_T1;
  const float* psum  = ws + WS_PS;
  const float* pmin  = ws + WS_PM; // nonneg float bits == float values
  const float* facc  = ws + WS_FA;

  __shared__ float sred[256];
  float s2 = 0.0f;
  for (int i = threadIdx.x; i < BB * MM; i += 256) s2 += pmin[i];
  sred[threadIdx.x] = s2;
  __syncthreads();
  for (int s = 128; s > 0; s >>= 1) {
    if (threadIdx.x < s) sred[threadIdx.x] += sred[threadIdx.x + s];
    __syncthreads();
  }
  if (threadIdx.x == 0) {
    float s1 = 0.0f;
    for (int b = 0; b < BB; ++b) s1 += t1num[b] / (psum[b] + 1e-6f);
    const float focal = -(facc[0] / facc[1]);
    out[0] = s1 * (1.0f / (float)BB) + sred[0] * (1.0f / (float)(BB * MM)) + focal;
  }
}

extern "C" void kernel_launch(void* const* d_in, const int* in_sizes, int n_in,
                              void* d_out, int out_size, void* d_ws, size_t ws_size,
                              hipStream_t stream) {
  const float* prob_map = (const float*)d_in[0]; // [8,256,256]
  const float* gt_map   = (const float*)d_in[1]; // [8,256,256]
  const float* gt       = (const float*)d_in[2]; // [8,128,2]
  const float* orig     = (const float*)d_in[3]; // [8,2]
  float* ws  = (float*)d_ws;
  float* out = (float*)d_out;

  init_ws_kernel<<<(WS_TOTAL + 255) / 256, 256, 0, stream>>>(ws);

  dim3 grid((NPIX / 16) / 8, BB); // 4096 pixel-tiles per batch, 8 waves/block
  whd_wmma_kernel<<<grid, 256, 0, stream>>>(
      prob_map, gt, orig, ws + WS_T1, ws + WS_PS,
      (unsigned int*)(ws + WS_PM));

  focal_kernel<<<(NPIX + 255) / 256, 256, 0, stream>>>(prob_map, gt_map,
                                                       ws + WS_FA);

  finalize_kernel<<<1, 256, 0, stream>>>(ws, out);
}